// QuarticBSplinePotential_81836306858493
// MI455X (gfx1250) — compile-verified
//
#include <hip/hip_runtime.h>

typedef float v2f __attribute__((ext_vector_type(2)));
typedef float v8f __attribute__((ext_vector_type(8)));

#define BOX_LO      (-3.0f)
#define SCALE_F     (0.1875f)          // 6/32, exact in binary
#define INV_SCALE_F (16.0f / 3.0f)

// Evaluate sum_j w[j] * S((x - c_j)/SCALE) using the 5-center support window.
// Within the window the bucket index is idx = 5 - p (p = window position), so
// the piecewise-quartic coefficients are compile-time constants. Fully
// branchless: out-of-range centers are nulled with a cndmask on the weight.
__device__ __forceinline__ float spline_window(float x, const float* __restrict__ w)
{
    const float C[5][5] = {
        // p=0 -> idx 5
        { 625.0f/(16.0f*24.0f), -125.0f/(2.0f*24.0f),  75.0f/(2.0f*24.0f), -10.0f/24.0f,  1.0f/24.0f },
        // p=1 -> idx 4
        {  55.0f/(4.0f*24.0f),     5.0f/24.0f,        -30.0f/24.0f,         20.0f/24.0f, -4.0f/24.0f },
        // p=2 -> idx 3
        { 115.0f/(8.0f*24.0f),     0.0f,              -15.0f/24.0f,          0.0f,        6.0f/24.0f },
        // p=3 -> idx 2
        {  55.0f/(4.0f*24.0f),    -5.0f/24.0f,        -30.0f/24.0f,        -20.0f/24.0f, -4.0f/24.0f },
        // p=4 -> idx 1
        { 625.0f/(16.0f*24.0f),  125.0f/(2.0f*24.0f),  75.0f/(2.0f*24.0f),  10.0f/24.0f,  1.0f/24.0f },
    };

    float u   = (x - BOX_LO) * INV_SCALE_F;      // x position in center-index units
    float j0f = ceilf(u - 2.5f);                 // first center in support window
    int   j0  = (int)j0f;
    float xs  = u - j0f;                         // xs for p=0; steps by -1 per p
    float acc = 0.0f;

#pragma unroll
    for (int p = 0; p < 5; ++p) {
        int   j  = j0 + p;
        int   jc = min(max(j, 0), 32);           // v_med3_i32
        float wv = w[jc];                        // ds_load_b32 (always in-bounds)
        float wj = (j == jc) ? wv : 0.0f;        // v_cndmask: zero if out of range
        float yp = fmaf(xs, C[p][4], C[p][3]);
        yp = fmaf(xs, yp, C[p][2]);
        yp = fmaf(xs, yp, C[p][1]);
        yp = fmaf(xs, yp, C[p][0]);
        acc = fmaf(yp, wj, acc);
        xs -= 1.0f;
    }
    return acc;
}

__global__ void qbsp_zero_kernel(float* __restrict__ out)
{
    if (threadIdx.x == 0 && blockIdx.x == 0) out[0] = 0.0f;
}

__global__ __launch_bounds__(256)
void QuarticBSplinePotential_81836306858493_kernel(const float* __restrict__ x,
                                                   const float* __restrict__ weights,
                                                   float* __restrict__ out,
                                                   long long n)
{
    __shared__ float w[33];
    __shared__ float wsum[8];

    const int tid = threadIdx.x;
    if (tid < 33) w[tid] = weights[tid];
    __syncthreads();

    long long base = ((long long)blockIdx.x * 256 + tid) * 4;
    float acc = 0.0f;
    if (base + 3 < n) {
        float4 v = *(const float4*)(x + base);      // global_load_b128
        acc += spline_window(v.x, w);
        acc += spline_window(v.y, w);
        acc += spline_window(v.z, w);
        acc += spline_window(v.w, w);
    } else {
        for (long long i = base; i < n; ++i) acc += spline_window(x[i], w);
    }

    // ---- wave-level reduction through the matrix engine (convergent point,
    // EXEC all-ones as WMMA requires).  A(16x4): lanes 0-15 feed column K=0,
    // lanes 16-31 feed column K=2 (VGPR0 of the pair); B = ones(4x16).
    // => D[m][n] = s[m] + s[m+16] for every n.
    v2f a; a.x = acc;  a.y = 0.0f;
    v2f b; b.x = 1.0f; b.y = 1.0f;
    v8f c = {};
    v8f d = __builtin_amdgcn_wmma_f32_16x16x4_f32(false, a, false, b,
                                                  (short)0, c, false, false);
    // per lane: sum of 8 D rows -> lanes 0-15 hold sum over indices {0..7,16..23},
    // lanes 16-31 hold sum over {8..15,24..31}
    float t = ((d[0] + d[1]) + (d[2] + d[3])) + ((d[4] + d[5]) + (d[6] + d[7]));
    t += __shfl_xor(t, 16, 32);                    // total across the wave

    if ((tid & 31) == 0) wsum[tid >> 5] = t;       // one slot per wave, no atomics
    __syncthreads();
    if (tid == 0) {
        float s = ((wsum[0] + wsum[1]) + (wsum[2] + wsum[3])) +
                  ((wsum[4] + wsum[5]) + (wsum[6] + wsum[7]));
        atomicAdd(out, s);                         // global_atomic_add_f32
    }
}

extern "C" void kernel_launch(void* const* d_in, const int* in_sizes, int n_in,
                              void* d_out, int out_size, void* d_ws, size_t ws_size,
                              hipStream_t stream)
{
    const float* x   = (const float*)d_in[0];
    const float* wts = (const float*)d_in[1];
    float*       out = (float*)d_out;
    long long    n   = (long long)in_sizes[0];

    qbsp_zero_kernel<<<1, 1, 0, stream>>>(out);

    const int threads = 256;
    long long elemsPerBlock = (long long)threads * 4;
    int blocks = (int)((n + elemsPerBlock - 1) / elemsPerBlock);
    if (blocks < 1) blocks = 1;

    QuarticBSplinePotential_81836306858493_kernel<<<blocks, threads, 0, stream>>>(
        x, wts, out, n);
}